// SASS_SS2D_39118562132163
// MI455X (gfx1250) — compile-verified
//
#include <hip/hip_runtime.h>
#include <hip/hip_bf16.h>
#include <math.h>

#define B_    4
#define Hh_   64
#define Ww_   64
#define L_    4096
#define DM_   96
#define DI_   192
#define KDIR_ 4
#define DST_  16
#define DTR_  6
#define NROW_ 38          // DTR + 2*DST
#define XPITCH_ 40        // padded row count for xdbl
#define PI_D  3.14159265358979323846

typedef float v2f __attribute__((ext_vector_type(2)));
typedef float v8f __attribute__((ext_vector_type(8)));

// ---------------------------------------------------------------------------
// K0a: radial sort keys (fp64, matches numpy argsort key computation)
// ---------------------------------------------------------------------------
__global__ void k_keys(double* __restrict__ keys) {
  int c = blockIdx.x * blockDim.x + threadIdx.x;
  if (c >= L_) return;
  int i = c >> 6, j = c & 63;
  double dy = (double)(i - 32), dx = (double)(j - 32);
  double r  = sqrt(dy * dy + dx * dx);
  double th = atan2(dy, dx);
  keys[c]       =  r * (double)(L_ + 1) + (th + PI_D) * 10.0;
  keys[L_ + c]  = -r * (double)(L_ + 1) + (th + PI_D) * 10.0;
}

// K0b: stable argsort by rank counting (ties broken by index, like np.argsort)
__global__ void k_radial_rank(const double* __restrict__ keys,
                              int* __restrict__ idx, int* __restrict__ inv) {
  int t = blockIdx.x * blockDim.x + threadIdx.x;
  if (t >= 2 * L_) return;
  int dir = t >> 12, c = t & (L_ - 1);
  const double* kk = keys + dir * L_;
  double my = kk[c];
  int rank = 0;
  for (int j = 0; j < L_; ++j) {
    double v = kk[j];
    rank += (v < my) || (v == my && j < c);
  }
  int d = 2 + dir;
  idx[d * L_ + rank] = c;
  inv[d * L_ + c]    = rank;
}

// K0c: diagonal / anti-diagonal snake orders (analytic position)
__global__ void k_snake(int* __restrict__ idx, int* __restrict__ inv) {
  int c = blockIdx.x * blockDim.x + threadIdx.x;
  if (c >= L_) return;
  int i = c >> 6, j = c & 63;
  for (int dir = 0; dir < 2; ++dir) {
    int s = (dir == 0) ? (i + j) : (i + (63 - j));
    int imin = max(0, s - 63), imax = min(63, s);
    int cum = 0;
    for (int t = 0; t < s; ++t)
      cum += min(63, t) + 1 - max(0, t - 63);
    int pos = (s & 1) ? (imax - i) : (i - imin);
    int p = cum + pos;
    idx[dir * L_ + p] = c;
    inv[dir * L_ + c] = p;
  }
}

// ---------------------------------------------------------------------------
// K1: in_proj GEMM  (16384x96)@(96x384)^T  via V_WMMA_F32_16X16X4_F32
// ---------------------------------------------------------------------------
__global__ void k_inproj(const float* __restrict__ x, const float* __restrict__ wp,
                         float* __restrict__ xcpre, float* __restrict__ z) {
  int wave = (blockIdx.x * blockDim.x + threadIdx.x) >> 5;
  int lane = threadIdx.x & 31;
  const int NT = (2 * DI_) / 16;          // 24
  int mt = wave / NT, nt = wave % NT;
  if (mt >= (B_ * L_) / 16) return;
  int row0 = mt * 16, col0 = nt * 16;
  int m = lane & 15, hi = lane >> 4;
  const float* arow = x  + (size_t)(row0 + m) * DM_;
  const float* brow = wp + (size_t)(col0 + m) * DM_;
  v8f acc = {};
  for (int kk = 0; kk < DM_; kk += 4) {
    v2f a, b;
    a.x = arow[kk + 2 * hi]; a.y = arow[kk + 2 * hi + 1];
    b.x = brow[kk + 2 * hi]; b.y = brow[kk + 2 * hi + 1];
    acc = __builtin_amdgcn_wmma_f32_16x16x4_f32(false, a, false, b,
                                                (short)0, acc, false, false);
  }
#pragma unroll
  for (int v = 0; v < 8; ++v) {
    int row = row0 + v + 8 * hi;
    int col = col0 + m;
    float val = acc[v];
    if (col < DI_) xcpre[(size_t)row * DI_ + col] = val;
    else           z[(size_t)row * DI_ + (col - DI_)] = val;
  }
}

// ---------------------------------------------------------------------------
// K2: depthwise 3x3 conv (SAME) + bias + SiLU; writes channel-major (B,DI,L)
// ---------------------------------------------------------------------------
__global__ void k_conv(const float* __restrict__ xcpre, const float* __restrict__ cw,
                       const float* __restrict__ cb, float* __restrict__ xcconv) {
  int t = blockIdx.x * blockDim.x + threadIdx.x;
  if (t >= B_ * L_ * DI_) return;
  int d = t % DI_; int rest = t / DI_;
  int j = rest & 63; rest >>= 6;
  int i = rest & 63; int b = rest >> 6;
  float s = cb[d];
#pragma unroll
  for (int di = -1; di <= 1; ++di)
#pragma unroll
    for (int dj = -1; dj <= 1; ++dj) {
      int ii = i + di, jj = j + dj;
      if (ii < 0 || ii > 63 || jj < 0 || jj > 63) continue;
      s += xcpre[((size_t)b * L_ + ii * 64 + jj) * DI_ + d] *
           cw[d * 9 + (di + 1) * 3 + (dj + 1)];
    }
  s = s / (1.f + expf(-s));
  xcconv[((size_t)b * DI_ + d) * L_ + i * 64 + j] = s;
}

// ---------------------------------------------------------------------------
// K3a: x_dbl = x_proj_weight[k] @ xs  (38x192 @ 192xL, gather over scan order)
// ---------------------------------------------------------------------------
__global__ void k_xdbl(const float* __restrict__ xcconv, const float* __restrict__ wx,
                       const int* __restrict__ idx, float* __restrict__ xdbl) {
  int wave = (blockIdx.x * blockDim.x + threadIdx.x) >> 5;
  int lane = threadIdx.x & 31;
  int t = wave;
  if (t >= B_ * KDIR_ * 3 * (L_ / 16)) return;
  int ntile = t % (L_ / 16); t /= (L_ / 16);
  int mtile = t % 3;         t /= 3;
  int k = t % KDIR_;  int b = t / KDIR_;
  int m0 = mtile * 16, l0 = ntile * 16;
  int m = lane & 15, hi = lane >> 4;
  int r = m0 + m;
  bool aval = (r < NROW_);
  const float* awx = wx + ((size_t)k * NROW_ + (aval ? r : 0)) * DI_;
  int g = idx[k * L_ + l0 + m];
  const float* bbase = xcconv + (size_t)b * DI_ * L_ + g;
  v8f acc = {};
  for (int kkk = 0; kkk < DI_; kkk += 4) {
    v2f a, bb;
    a.x = aval ? awx[kkk + 2 * hi]     : 0.f;
    a.y = aval ? awx[kkk + 2 * hi + 1] : 0.f;
    bb.x = bbase[(size_t)(kkk + 2 * hi) * L_];
    bb.y = bbase[(size_t)(kkk + 2 * hi + 1) * L_];
    acc = __builtin_amdgcn_wmma_f32_16x16x4_f32(false, a, false, bb,
                                                (short)0, acc, false, false);
  }
#pragma unroll
  for (int v = 0; v < 8; ++v) {
    int rr = m0 + v + 8 * hi;
    if (rr < NROW_)
      xdbl[(((size_t)(b * KDIR_ + k) * XPITCH_) + rr) * L_ + l0 + m] = acc[v];
  }
}

// K3b: Bs/Cs extraction (+ direction_Bs), layout (b,k,l,32)
__global__ void k_bsc(const float* __restrict__ xdbl, const float* __restrict__ dirB,
                      float* __restrict__ bsc) {
  int t = blockIdx.x * blockDim.x + threadIdx.x;
  if (t >= B_ * KDIR_ * L_ * 2 * DST_) return;
  int l = t & (L_ - 1);
  int c32 = (t >> 12) & 31;
  int bk = t >> 17;
  int k = bk & 3;
  float v = xdbl[((size_t)bk * XPITCH_ + DTR_ + c32) * L_ + l];
  if (c32 < DST_) v += dirB[k * DST_ + c32];
  bsc[((size_t)bk * L_ + l) * 32 + c32] = v;
}

// K3c: dt projection (K=6) + bias + softplus, layout (b,k,d,l)
__global__ void k_delta(const float* __restrict__ xdbl, const float* __restrict__ wdt,
                        const float* __restrict__ bias, float* __restrict__ delta) {
  int t = blockIdx.x * blockDim.x + threadIdx.x;
  if (t >= B_ * KDIR_ * DI_ * L_) return;
  int l = t & (L_ - 1);
  int d = (t >> 12) % DI_;
  int bk = t / (DI_ * L_);
  int k = bk & 3;
  float s = bias[k * DI_ + d];
#pragma unroll
  for (int rr = 0; rr < DTR_; ++rr)
    s += wdt[((size_t)k * DI_ + d) * DTR_ + rr] * xdbl[((size_t)bk * XPITCH_ + rr) * L_ + l];
  float dl = (s > 20.f) ? s : log1pf(expf(s));
  delta[((size_t)bk * DI_ + d) * L_ + l] = dl;
}

// ---------------------------------------------------------------------------
// K4: selective scan — wave handles 2 channels, lane-per-state (16 states)
// ---------------------------------------------------------------------------
__global__ void k_scan(const float* __restrict__ delta, const float* __restrict__ xcconv,
                       const float* __restrict__ bsc,   const int* __restrict__ idx,
                       const float* __restrict__ Alogs, const float* __restrict__ Ds,
                       float* __restrict__ outy) {
  int wave = (blockIdx.x * blockDim.x + threadIdx.x) >> 5;
  int lane = threadIdx.x & 31;
  int half = lane >> 4, n = lane & 15;
  int c = wave * 2 + half;                 // channel = (b*K+k)*192 + d
  if (c >= B_ * KDIR_ * DI_) return;
  int d = c % DI_, bk = c / DI_;
  int k = bk & 3, b = bk >> 2;
  const float* dptr  = delta  + (size_t)c * L_;
  const float* ubase = xcconv + ((size_t)b * DI_ + d) * L_;
  const float* bscp  = bsc    + (size_t)bk * L_ * 32;
  const int*   gptr  = idx    + k * L_;
  float Dval = Ds[k * DI_ + d];
  float an = -expf(Alogs[((size_t)(k * DI_ + d)) * DST_ + n]);   // = -(n+1)
  float h = 0.f;
  float* oy = outy + (size_t)c * L_;
  for (int l = 0; l < L_; ++l) {
    __builtin_prefetch(dptr + l + 16, 0, 1);
    __builtin_prefetch(bscp + (size_t)(l + 16) * 32 + n, 0, 1);
    float dt = dptr[l];
    int   g  = gptr[l];
    float u  = ubase[g];
    float Bn = bscp[(size_t)l * 32 + n];
    float Cn = bscp[(size_t)l * 32 + 16 + n];
    float e = expf(dt * an);
    h = h * e + dt * u * Bn;
    float y = h * Cn;
    y += __shfl_xor(y, 1, 16);
    y += __shfl_xor(y, 2, 16);
    y += __shfl_xor(y, 4, 16);
    y += __shfl_xor(y, 8, 16);
    if (n == 0) oy[l] = y + Dval * u;
  }
}

// ---------------------------------------------------------------------------
// K5: per-pixel fusion weights (2->32 GELU -> 4, softmax)
// ---------------------------------------------------------------------------
__global__ void k_fusew(const float* __restrict__ w1, const float* __restrict__ b1,
                        const float* __restrict__ w2, const float* __restrict__ b2,
                        float* __restrict__ w4) {
  int l = blockIdx.x * blockDim.x + threadIdx.x;
  if (l >= L_) return;
  int i = l >> 6, j = l & 63;
  float dy = ((float)i - 32.f) / 32.f;
  float dx = ((float)j - 32.f) / 32.f;
  float r = sqrtf(dy * dy + dx * dx); if (r > 2.f) r = 2.f;
  float th = atan2f(dy, dx) * (float)(1.0 / PI_D);
  float logits[4] = { b2[0], b2[1], b2[2], b2[3] };
  for (int hh = 0; hh < 32; ++hh) {
    float hv = r * w1[hh * 2 + 0] + th * w1[hh * 2 + 1] + b1[hh];
    float gg = 0.5f * hv * (1.f + erff(hv * 0.7071067811865476f));
#pragma unroll
    for (int kk = 0; kk < 4; ++kk) logits[kk] += w2[kk * 32 + hh] * gg;
  }
  float mx = fmaxf(fmaxf(logits[0], logits[1]), fmaxf(logits[2], logits[3]));
  float ex[4], se = 0.f;
#pragma unroll
  for (int kk = 0; kk < 4; ++kk) { ex[kk] = expf(logits[kk] - mx); se += ex[kk]; }
#pragma unroll
  for (int kk = 0; kk < 4; ++kk) w4[l * 4 + kk] = ex[kk] / se;
}

// ---------------------------------------------------------------------------
// K6a: unscatter + direction fuse + LayerNorm + SiLU gate (wave per (b,l))
// ---------------------------------------------------------------------------
__global__ void k_norm(const float* __restrict__ outy, const int* __restrict__ inv,
                       const float* __restrict__ w4,   const float* __restrict__ z,
                       const float* __restrict__ lng,  const float* __restrict__ lnb,
                       float* __restrict__ yact) {
  int wave = (blockIdx.x * blockDim.x + threadIdx.x) >> 5;
  int lane = threadIdx.x & 31;
  if (wave >= B_ * L_) return;
  int b = wave >> 12, l = wave & (L_ - 1);
  int invk[4]; float wk[4];
#pragma unroll
  for (int k = 0; k < 4; ++k) { invk[k] = inv[k * L_ + l]; wk[k] = w4[l * 4 + k]; }
  float yf[6], sum = 0.f, sq = 0.f;
#pragma unroll
  for (int t = 0; t < 6; ++t) {
    int d = lane + 32 * t;
    float acc = 0.f;
#pragma unroll
    for (int k = 0; k < 4; ++k)
      acc += wk[k] * outy[(((size_t)(b * KDIR_ + k) * DI_) + d) * L_ + invk[k]];
    yf[t] = acc; sum += acc; sq += acc * acc;
  }
  for (int m = 1; m < 32; m <<= 1) {
    sum += __shfl_xor(sum, m, 32);
    sq  += __shfl_xor(sq,  m, 32);
  }
  float mean = sum / (float)DI_;
  float var  = sq / (float)DI_ - mean * mean;
  float rstd = rsqrtf(var + 1e-5f);
#pragma unroll
  for (int t = 0; t < 6; ++t) {
    int d = lane + 32 * t;
    float yn = (yf[t] - mean) * rstd * lng[d] + lnb[d];
    float zv = z[((size_t)b * L_ + l) * DI_ + d];
    yact[((size_t)b * L_ + l) * DI_ + d] = yn * (zv / (1.f + expf(-zv)));
  }
}

// ---------------------------------------------------------------------------
// K6b: out_proj GEMM (16384x192)@(192x96)^T via WMMA
// ---------------------------------------------------------------------------
__global__ void k_outproj(const float* __restrict__ yact, const float* __restrict__ wo,
                          float* __restrict__ out) {
  int wave = (blockIdx.x * blockDim.x + threadIdx.x) >> 5;
  int lane = threadIdx.x & 31;
  const int NT = DM_ / 16;                 // 6
  int mt = wave / NT, nt = wave % NT;
  if (mt >= (B_ * L_) / 16) return;
  int row0 = mt * 16, col0 = nt * 16;
  int m = lane & 15, hi = lane >> 4;
  const float* arow = yact + (size_t)(row0 + m) * DI_;
  const float* brow = wo   + (size_t)(col0 + m) * DI_;
  v8f acc = {};
  for (int kk = 0; kk < DI_; kk += 4) {
    v2f a, b;
    a.x = arow[kk + 2 * hi]; a.y = arow[kk + 2 * hi + 1];
    b.x = brow[kk + 2 * hi]; b.y = brow[kk + 2 * hi + 1];
    acc = __builtin_amdgcn_wmma_f32_16x16x4_f32(false, a, false, b,
                                                (short)0, acc, false, false);
  }
#pragma unroll
  for (int v = 0; v < 8; ++v)
    out[(size_t)(row0 + v + 8 * hi) * DM_ + col0 + m] = acc[v];
}

// ---------------------------------------------------------------------------
extern "C" void kernel_launch(void* const* d_in, const int* in_sizes, int n_in,
                              void* d_out, int out_size, void* d_ws, size_t ws_size,
                              hipStream_t stream) {
  const float* x      = (const float*)d_in[0];
  const float* inproj = (const float*)d_in[1];
  const float* convw  = (const float*)d_in[2];
  const float* convb  = (const float*)d_in[3];
  const float* xprojw = (const float*)d_in[4];
  const float* dtw    = (const float*)d_in[5];
  const float* dtb    = (const float*)d_in[6];
  const float* Alogs  = (const float*)d_in[7];
  const float* Ds     = (const float*)d_in[8];
  const float* dirB   = (const float*)d_in[9];
  const float* pfw1   = (const float*)d_in[10];
  const float* pfb1   = (const float*)d_in[11];
  const float* pfw2   = (const float*)d_in[12];
  const float* pfb2   = (const float*)d_in[13];
  const float* lng    = (const float*)d_in[14];
  const float* lnb    = (const float*)d_in[15];
  const float* outw   = (const float*)d_in[16];
  float* out = (float*)d_out;

  char* ws = (char*)d_ws;
  size_t off = 0;
  int*    idx   = (int*)(ws + off);     off += (size_t)KDIR_ * L_ * 4;       // 64 KB
  int*    inv   = (int*)(ws + off);     off += (size_t)KDIR_ * L_ * 4;       // 64 KB
  double* keys  = (double*)(ws + off);  off += (size_t)2 * L_ * 8;           // 64 KB
  float*  xcpre = (float*)(ws + off);   off += (size_t)B_ * L_ * DI_ * 4;    // 12 MB (reused as yact)
  float*  zbuf  = (float*)(ws + off);   off += (size_t)B_ * L_ * DI_ * 4;    // 12 MB
  float*  xcconv= (float*)(ws + off);   off += (size_t)B_ * DI_ * L_ * 4;    // 12 MB
  float*  xdbl  = (float*)(ws + off);   off += (size_t)B_ * KDIR_ * XPITCH_ * L_ * 4; // 10 MB
  float*  bsc   = (float*)(ws + off);   off += (size_t)B_ * KDIR_ * L_ * 32 * 4;      // 8 MB
  float*  delta = (float*)(ws + off);   off += (size_t)B_ * KDIR_ * DI_ * L_ * 4;     // 48 MB
  float*  outy  = (float*)(ws + off);   off += (size_t)B_ * KDIR_ * DI_ * L_ * 4;     // 48 MB
  float*  w4    = (float*)(ws + off);   off += (size_t)L_ * 4 * 4;           // 64 KB
  float*  yact  = xcpre;                // alias: xcpre dead after conv

  const int TB = 256;
  // scan orders
  k_keys<<<(L_ + TB - 1) / TB, TB, 0, stream>>>(keys);
  k_radial_rank<<<(2 * L_ + TB - 1) / TB, TB, 0, stream>>>(keys, idx, inv);
  k_snake<<<(L_ + TB - 1) / TB, TB, 0, stream>>>(idx, inv);
  // in_proj (WMMA): 1024*24 waves
  k_inproj<<<(1024 * 24) / 8, TB, 0, stream>>>(x, inproj, xcpre, zbuf);
  // depthwise conv + SiLU
  k_conv<<<(B_ * L_ * DI_ + TB - 1) / TB, TB, 0, stream>>>(xcpre, convw, convb, xcconv);
  // x_dbl (WMMA, gathered): 16*3*256 waves
  k_xdbl<<<(B_ * KDIR_ * 3 * (L_ / 16)) / 8, TB, 0, stream>>>(xcconv, xprojw, idx, xdbl);
  k_bsc<<<(B_ * KDIR_ * L_ * 32 + TB - 1) / TB, TB, 0, stream>>>(xdbl, dirB, bsc);
  k_delta<<<(B_ * KDIR_ * DI_ * L_ + TB - 1) / TB, TB, 0, stream>>>(xdbl, dtw, dtb, delta);
  // selective scan: 1536 waves (2 channels each)
  k_scan<<<(B_ * KDIR_ * DI_ / 2) / 8, TB, 0, stream>>>(delta, xcconv, bsc, idx, Alogs, Ds, outy);
  // fusion weights + norm + gate
  k_fusew<<<(L_ + TB - 1) / TB, TB, 0, stream>>>(pfw1, pfb1, pfw2, pfb2, w4);
  k_norm<<<(B_ * L_) / 8, TB, 0, stream>>>(outy, inv, w4, zbuf, lng, lnb, yact);
  // out_proj (WMMA): 1024*6 waves
  k_outproj<<<(1024 * 6) / 8, TB, 0, stream>>>(yact, outw, out);
  (void)in_sizes; (void)n_in; (void)out_size; (void)ws_size; (void)off;
}